// SelfCorrelationComputation_33285996544487
// MI455X (gfx1250) — compile-verified
//
#include <hip/hip_runtime.h>

// SelfCorrelationComputation for q:(16,64,32,32) f32.
// out[b,c,h,w,u,v] = qn[b,c,h+u-2,w+v-2] * qn[b,c,h,w]; qn = q / max(||q||_c, 1e-12)
//
// Memory-bound: 4.2 MB in, 104.9 MB out -> ~4.7 us floor @ 23.3 TB/s.
// 25x amplification happens in LDS; input planes arrive via gfx1250 async
// global->LDS copies (ASYNCcnt); output leaves as coalesced 128-bit NT stores.

typedef float f32x4 __attribute__((ext_vector_type(4)));

#define SC_EPS   1e-12f
#define SC_C     64
#define SC_HW    1024      // 32*32
#define SC_PADW  36        // 32 + 2*2 zero pad

// ---------------------------------------------------------------------------
// Pass 2 (emitted first so the disasm snippet shows the async/NT paths):
// one workgroup per (b,c) plane. grid = B*64, block = 256.
// ---------------------------------------------------------------------------
template <bool USE_WS>
__global__ void __launch_bounds__(256)
sc_main_kernel(const float* __restrict__ q,
               const float* __restrict__ den_ws,
               float* __restrict__ out) {
    __shared__ float lds_qn[SC_PADW * SC_PADW]; // zero-padded normalized plane
    __shared__ f32x4 lds_stage4[1600];          // 256 pixels * 25 taps staging
    __shared__ f32x4 lds_q4[256];               // async dst: raw q plane (4KB)
    __shared__ f32x4 lds_d4[256];               // async dst: denominators (4KB)

    const int bc = blockIdx.x;     // b*64 + c
    const int b  = bc >> 6;
    const int t  = threadIdx.x;

    // Kick off the async global->LDS copies FIRST so they overlap the
    // zero-fill below. Each lane moves 16B; 256 threads cover the whole
    // 4KB plane + 4KB of denominators. Tracked by ASYNCcnt.
    if (USE_WS) {
        unsigned long long gq = (unsigned long long)(q      + ((size_t)bc << 10) + (t << 2));
        unsigned long long gd = (unsigned long long)(den_ws + ((size_t)b  << 10) + (t << 2));
        unsigned lq = (unsigned)(size_t)(&lds_q4[t]);   // low 32 bits = LDS offset
        unsigned ld = (unsigned)(size_t)(&lds_d4[t]);
        asm volatile("global_load_async_to_lds_b128 %0, %1, off"
                     :: "v"(lq), "v"(gq) : "memory");
        asm volatile("global_load_async_to_lds_b128 %0, %1, off"
                     :: "v"(ld), "v"(gd) : "memory");
    }

    // Zero the padded tile (borders stay zero == reference zero padding).
    for (int i = t; i < SC_PADW * SC_PADW; i += 256) lds_qn[i] = 0.0f;

    f32x4 qn4 = {0.f, 0.f, 0.f, 0.f};
    if (USE_WS) {
        asm volatile("s_wait_asynccnt 0x0" ::: "memory");
    } else {
        // Fallback (no workspace): recompute norm from L2-resident input.
        f32x4 acc = {0.f, 0.f, 0.f, 0.f};
        const f32x4* qv = (const f32x4*)q;
#pragma unroll 8
        for (int c = 0; c < SC_C; ++c) {
            f32x4 x = qv[((size_t)(b * SC_C + c) << 8) + t];
            acc += x * x;
        }
        f32x4 mine = qv[((size_t)bc << 8) + t];
#pragma unroll
        for (int j = 0; j < 4; ++j) qn4[j] = mine[j] / fmaxf(sqrtf(acc[j]), SC_EPS);
    }
    __syncthreads();  // zero-fill + async LDS data visible to all waves

    // Scatter normalized values into the padded tile interior.
    {
        const int p = t << 2;              // pixel base (w is a multiple of 4)
        const int h = p >> 5, w = p & 31;
        float* dst = &lds_qn[(h + 2) * SC_PADW + (w + 2)];
        if (USE_WS) {
            f32x4 xq = lds_q4[t];
            f32x4 xd = lds_d4[t];
#pragma unroll
            for (int j = 0; j < 4; ++j) dst[j] = xq[j] / xd[j];
        } else {
#pragma unroll
            for (int j = 0; j < 4; ++j) dst[j] = qn4[j];
        }
    }
    __syncthreads();

    // 25x amplification: all tap offsets are compile-time constants,
    // stage per 256-pixel chunk, then stream out coalesced b128 NT stores.
    float* stage = (float*)lds_stage4;
    for (int chunk = 0; chunk < 4; ++chunk) {
        const int p    = (chunk << 8) + t;
        const int h    = p >> 5, w = p & 31;
        const int base = h * SC_PADW + w;
        const float center = lds_qn[base + 2 * SC_PADW + 2];
        float* st = stage + t * 25;
#pragma unroll
        for (int u = 0; u < 5; ++u)
#pragma unroll
            for (int v = 0; v < 5; ++v)
                st[u * 5 + v] = center * lds_qn[base + u * SC_PADW + v];
        __syncthreads();

        // 1600 f32x4 = 6 full rounds of 256 threads + 64-thread tail.
        f32x4* dst4 = (f32x4*)(out + (size_t)bc * 25600 + (size_t)chunk * 6400);
#pragma unroll
        for (int k = 0; k < 6; ++k) {
            const int i = k * 256 + t;
            __builtin_nontemporal_store(lds_stage4[i], &dst4[i]);
        }
        if (t < 64) {
            const int i = 1536 + t;
            __builtin_nontemporal_store(lds_stage4[i], &dst4[i]);
        }
        __syncthreads();
    }
}

// ---------------------------------------------------------------------------
// Pass 1: per-pixel channel L2 norm denominators -> d_ws  (B*1024 floats)
// grid = B, block = 256; thread t owns pixels [4t, 4t+4) of batch b.
// ---------------------------------------------------------------------------
__global__ void __launch_bounds__(256)
sc_norm_kernel(const float* __restrict__ q, float* __restrict__ den_ws) {
    const int b = blockIdx.x;
    const int t = threadIdx.x;
    const f32x4* qv = (const f32x4*)q;
    f32x4 acc = {0.f, 0.f, 0.f, 0.f};
#pragma unroll 8
    for (int c = 0; c < SC_C; ++c) {
        f32x4 x = qv[((size_t)(b * SC_C + c) << 8) + t];
        acc += x * x;
    }
    f32x4 r;
#pragma unroll
    for (int j = 0; j < 4; ++j) r[j] = fmaxf(sqrtf(acc[j]), SC_EPS);
    ((f32x4*)den_ws)[((size_t)b << 8) + t] = r;
}

// ---------------------------------------------------------------------------
extern "C" void kernel_launch(void* const* d_in, const int* in_sizes, int n_in,
                              void* d_out, int out_size, void* d_ws, size_t ws_size,
                              hipStream_t stream) {
    (void)n_in; (void)out_size;
    const float* q   = (const float*)d_in[0];
    float*       out = (float*)d_out;
    float*       ws  = (float*)d_ws;

    const int Bn = in_sizes[0] / (SC_C * SC_HW);   // 16 for the reference shapes
    const bool use_ws = (ws != nullptr) &&
                        (ws_size >= (size_t)Bn * SC_HW * sizeof(float));

    if (use_ws) {
        sc_norm_kernel<<<Bn, 256, 0, stream>>>(q, ws);
        sc_main_kernel<true><<<Bn * SC_C, 256, 0, stream>>>(q, ws, out);
    } else {
        sc_main_kernel<false><<<Bn * SC_C, 256, 0, stream>>>(q, nullptr, out);
    }
}